// ModalSynthFreqs_59030030516773
// MI455X (gfx1250) — compile-verified
//
#include <hip/hip_runtime.h>
#include <math.h>

// ModalSynthFreqs for MI455X (gfx1250, wave32).
// B=16 batches, M=64 modes, F=189 knots, HOP=512, n=(F-1)*HOP=96256.
// Stage 1: per-(b,m) block phase bases (f64 running sum -> mod 2pi, f32 store).
// Stage 2: per (b, block) workgroup; intra-block inclusive prefix sum done as
//          P = L * X via chained V_WMMA_F32_16X16X4_F32 (L = lower-tri ones),
//          then __cosf / Hann-amp / mode reduction.

#define F_KNOTS 189
#define HOP     512
#define NSAMP   96256
#define NBLK    188
#define BATCH   16
#define MODES   64
#define SCALEF  (189.0f / 96256.0f)
#define TWO_PI_D 6.283185307179586476925286766559
#define PI_F    3.14159265358979323846f

typedef float v2f __attribute__((ext_vector_type(2)));
typedef float v8f __attribute__((ext_vector_type(8)));

__device__ __forceinline__ float f0_from_row(const float* row, int i0, float fr) {
    int i1 = (i0 < F_KNOTS - 1) ? i0 + 1 : F_KNOTS - 1;
    float f = row[i0] * (1.0f - fr) + row[i1] * fr;
    return fminf(fmaxf(f, 0.0f), PI_F);
}

// ---------------------------------------------------------------------------
// Stage 1: one wave (32 lanes) per (b,m) row. Each block of 512 samples:
// lane sums 16 consecutive samples, shfl_xor tree-reduce, f64 running prefix.
// base[row][blk] = (phase + sum of f0 before block blk) mod 2pi.
// ---------------------------------------------------------------------------
__global__ void __launch_bounds__(32)
modal_stage1(const float* __restrict__ freq, const float* __restrict__ phase,
             float* __restrict__ base) {
    const int pair = blockIdx.x;          // b*MODES + m
    const int lane = threadIdx.x;

    __shared__ float row[F_KNOTS + 3];
    __builtin_prefetch(&freq[pair * F_KNOTS], 0, 1);
    for (int k = lane; k < F_KNOTS; k += 32) row[k] = freq[pair * F_KNOTS + k];
    __syncthreads();

    double running = fmod((double)phase[pair], TWO_PI_D);

    for (int blk = 0; blk < NBLK; ++blk) {
        float s = 0.0f;
        const int tb = blk * HOP + lane * 16;
#pragma unroll
        for (int i = 0; i < 16; ++i) {
            int t = tb + i;
            float p = fminf(fmaxf(((float)t + 0.5f) * SCALEF - 0.5f, 0.0f),
                            (float)(F_KNOTS - 1));
            int i0 = (int)p;
            float fr = p - (float)i0;
            s += f0_from_row(row, i0, fr);
        }
#pragma unroll
        for (int d = 16; d >= 1; d >>= 1) s += __shfl_xor(s, d, 32);
        if (lane == 0) base[pair * NBLK + blk] = (float)fmod(running, TWO_PI_D);
        running += (double)s;
    }
}

// ---------------------------------------------------------------------------
// Stage 2: one workgroup (256 threads = 8 waves) per (b, blk).
// Each wave handles 8 modes. For each mode:
//   X[k][tile] = f0 at t = tile*16 + k  (32 tiles of 16 samples = 512)
//   P = L * X  via 4x V_WMMA_F32_16X16X4_F32 per 16-tile group (2 groups)
//   tile sums -> 32-lane shfl_up scan -> per-tile offsets
//   phase = base + offset + P ; contrib = __cosf(phase) * hann_amp
// Waves merge partial sums through LDS atomics.
//
// WMMA f32 16x16x4 layouts (ISA 7.12.2):
//   A (16x4): lane 0-15 M=lane, lanes 16-31 M=lane-16; K = vgpr + 2*(lane>=16)
//   B (4x16): N = lane&15; K = vgpr + 2*(lane>=16)   (mirrors A)
//   C/D (16x16): N = lane&15; M = vgpr + 8*(lane>=16)
// ---------------------------------------------------------------------------
__global__ void __launch_bounds__(256)
modal_stage2(const float* __restrict__ amp, const float* __restrict__ freq,
             const float* __restrict__ base, float* __restrict__ out) {
    const int b   = blockIdx.x / NBLK;
    const int blk = blockIdx.x % NBLK;
    const int tid  = threadIdx.x;
    const int lane = tid & 31;
    const int w    = tid >> 5;          // wave id 0..7

    __shared__ float s_frac[HOP];
    __shared__ int   s_i0[HOP];
    __shared__ float s_wlo[HOP];
    __shared__ float s_whi[HOP];
    __shared__ float s_acc[HOP];
    __shared__ float s_row[8][F_KNOTS + 3];
    __shared__ float s_ts[8][32];
    __shared__ float s_off[8][32];

    // Per-block precompute: interpolation indices/fracs and Hann window halves.
    for (int r = tid; r < HOP; r += 256) {
        int t = blk * HOP + r;
        float p = fminf(fmaxf(((float)t + 0.5f) * SCALEF - 0.5f, 0.0f),
                        (float)(F_KNOTS - 1));
        int i0 = (int)p;
        s_i0[r]   = i0;
        s_frac[r] = p - (float)i0;
        float cw = __cosf(0.006135923151542565f * (float)r);  // cos(2*pi*r/1024)
        s_wlo[r] = 0.5f - 0.5f * cw;   // win[r]
        s_whi[r] = 0.5f + 0.5f * cw;   // win[r+512]
        s_acc[r] = 0.0f;
    }
    __syncthreads();

    const int Mrow  = lane & 15;
    const int half  = lane >> 4;
    const int khalf = half * 2;

    v8f acc0 = {};
    v8f acc1 = {};

    for (int i = 0; i < 8; ++i) {
        const int m = w * 8 + i;
        const int rowbase = (b * MODES + m) * F_KNOTS;
        __builtin_prefetch(&freq[rowbase], 0, 1);
        for (int k = lane; k < F_KNOTS; k += 32)
            s_row[w][k] = freq[rowbase + k];

        const float a0 = amp[rowbase + blk];       // frame q   (win upper half)
        const float a1 = amp[rowbase + blk + 1];   // frame q+1 (win lower half)
        const float bphase = base[(b * MODES + m) * NBLK + blk];

        v8f P[2];
#pragma unroll
        for (int g = 0; g < 2; ++g) {
            const int tile = g * 16 + Mrow;        // column index (N)
            v8f c = {};
#pragma unroll
            for (int j = 0; j < 4; ++j) {
                const int kb  = 4 * j + khalf;
                const int tl0 = tile * 16 + kb;    // in-block sample index
                v2f Bv;
                Bv.x = f0_from_row(s_row[w], s_i0[tl0],     s_frac[tl0]);
                Bv.y = f0_from_row(s_row[w], s_i0[tl0 + 1], s_frac[tl0 + 1]);
                v2f Av;                             // lower-triangular ones
                Av.x = (kb     <= Mrow) ? 1.0f : 0.0f;
                Av.y = (kb + 1 <= Mrow) ? 1.0f : 0.0f;
                c = __builtin_amdgcn_wmma_f32_16x16x4_f32(
                        false, Av, false, Bv, (short)0, c, false, false);
            }
            P[g] = c;
            if (half) s_ts[w][g * 16 + (lane - 16)] = c[7];  // M=15 row = tile sum
        }

        // Exclusive scan of 32 tile sums across the wave.
        float ts = s_ts[w][lane];
        float incl = ts;
#pragma unroll
        for (int d = 1; d < 32; d <<= 1) {
            float o = __shfl_up(incl, d, 32);
            if (lane >= d) incl += o;
        }
        s_off[w][lane] = incl - ts;

#pragma unroll
        for (int g = 0; g < 2; ++g) {
            const int tile = g * 16 + Mrow;
            const float off = s_off[w][tile];
#pragma unroll
            for (int v = 0; v < 8; ++v) {
                const int r = tile * 16 + v + 8 * half;  // D: M = v + 8*half
                const float ph = bphase + off + P[g][v];
                const float a  = a0 * s_whi[r] + a1 * s_wlo[r];
                // |ph| <= 2pi + 512*pi ~ 1.6e3: fast v_cos path is accurate to
                // ~1e-4 rad here, far below the reference's f32-cumsum noise.
                const float contrib = __cosf(ph) * a;
                if (g == 0) acc0[v] += contrib; else acc1[v] += contrib;
            }
        }
    }

    // Merge the 8 waves' partial mode-sums.
#pragma unroll
    for (int v = 0; v < 8; ++v) {
        const int r0 = (0 * 16 + Mrow) * 16 + v + 8 * half;   // samples   0..255
        const int r1 = (1 * 16 + Mrow) * 16 + v + 8 * half;   // samples 256..511
        atomicAdd(&s_acc[r0], acc0[v]);
        atomicAdd(&s_acc[r1], acc1[v]);
    }
    __syncthreads();

    for (int r = tid; r < HOP; r += 256)
        out[b * NSAMP + blk * HOP + r] = s_acc[r];
}

// ---------------------------------------------------------------------------
extern "C" void kernel_launch(void* const* d_in, const int* in_sizes, int n_in,
                              void* d_out, int out_size, void* d_ws, size_t ws_size,
                              hipStream_t stream) {
    (void)in_sizes; (void)n_in; (void)out_size; (void)ws_size;
    const float* amp  = (const float*)d_in[0];   // [16,64,189]
    const float* freq = (const float*)d_in[1];   // [16,64,189]
    const float* phs  = (const float*)d_in[2];   // [16,64]
    float* out  = (float*)d_out;                 // [16,96256]
    float* base = (float*)d_ws;                  // 1024*188 floats = 770 KB

    modal_stage1<<<BATCH * MODES, 32, 0, stream>>>(freq, phs, base);
    modal_stage2<<<BATCH * NBLK, 256, 0, stream>>>(amp, freq, base, out);
}